// GNN_55473797595402
// MI455X (gfx1250) — compile-verified
//
#include <hip/hip_runtime.h>
#include <hip/hip_bf16.h>

typedef __attribute__((ext_vector_type(2))) float v2f;
typedef __attribute__((ext_vector_type(8))) float v8f;

#define N_NODES   50000
#define N_EDGES   1600000
#define IN_FEATS  256
#define OUT_FEATS 64
#define M_TILES   (N_NODES / 16)      // 3125, exact
#define W_STRIDE  260                 // 256 + 4 pad: bank-conflict-free B reads

// ---------------- zero out + deg ----------------
__global__ void GNN_zero_kernel(float* __restrict__ out, float* __restrict__ deg) {
    int i = blockIdx.x * blockDim.x + threadIdx.x;
    if (i < N_NODES * OUT_FEATS) out[i] = 0.0f;
    if (i < N_NODES)             deg[i] = 0.0f;
}

// ---------------- degree accumulation ----------------
__global__ void GNN_deg_kernel(const int* __restrict__ edge_dst, float* __restrict__ deg) {
    int e = blockIdx.x * blockDim.x + threadIdx.x;
    if (e < N_EDGES) atomicAdd(&deg[edge_dst[e]], 1.0f);
}

// ---------------- h = relu(x @ W^T + b) via v_wmma_f32_16x16x4_f32 ----------------
__launch_bounds__(256)
__global__ void GNN_gemm_relu_kernel(const float* __restrict__ x,
                                     const float* __restrict__ W,
                                     const float* __restrict__ b,
                                     float* __restrict__ h) {
    __shared__ float Wlds[OUT_FEATS * W_STRIDE];   // 66,560 B (LDS is 320 KB/WGP)

    // Stage W [64 x 256] row-major into padded LDS
    for (int i = threadIdx.x; i < OUT_FEATS * IN_FEATS; i += 256) {
        int n = i >> 8;          // output feature (row of W)
        int k = i & 255;         // input feature
        Wlds[n * W_STRIDE + k] = W[i];
    }
    __syncthreads();

    const int lane  = threadIdx.x & 31;
    const int wave  = threadIdx.x >> 5;
    const int tile  = blockIdx.x * 8 + wave;       // M-tile index
    if (tile >= M_TILES) return;

    const int row0  = tile * 16;
    const int nlane = lane & 15;
    const int koff  = (lane >> 4) * 2;             // 0 for lanes 0-15, 2 for 16-31
    const int mrow  = row0 + nlane;                // A-matrix row owned by this lane

    const float* xrow = x + (size_t)mrow * IN_FEATS + koff;
    const float* b0p  = &Wlds[( 0 + nlane) * W_STRIDE + koff];
    const float* b1p  = &Wlds[(16 + nlane) * W_STRIDE + koff];
    const float* b2p  = &Wlds[(32 + nlane) * W_STRIDE + koff];
    const float* b3p  = &Wlds[(48 + nlane) * W_STRIDE + koff];

    v8f c0 = {}, c1 = {}, c2 = {}, c3 = {};

    #pragma unroll 4
    for (int k = 0; k < IN_FEATS; k += 4) {
        // A 16x4 fp32: lanes 0-15 hold K=k..k+1, lanes 16-31 hold K=k+2..k+3
        v2f a  = *(const v2f*)(xrow + k);
        // B 4x16 fp32 per N-tile, same K split (B = W^T, i.e. b[k][n] = W[n][k])
        v2f bb0 = *(const v2f*)(b0p + k);
        v2f bb1 = *(const v2f*)(b1p + k);
        v2f bb2 = *(const v2f*)(b2p + k);
        v2f bb3 = *(const v2f*)(b3p + k);
        c0 = __builtin_amdgcn_wmma_f32_16x16x4_f32(false, a, false, bb0, (short)0, c0, false, false);
        c1 = __builtin_amdgcn_wmma_f32_16x16x4_f32(false, a, false, bb1, (short)0, c1, false, false);
        c2 = __builtin_amdgcn_wmma_f32_16x16x4_f32(false, a, false, bb2, (short)0, c2, false, false);
        c3 = __builtin_amdgcn_wmma_f32_16x16x4_f32(false, a, false, bb3, (short)0, c3, false, false);
    }

    // C/D 16x16 fp32 layout: VGPR v -> M = v (lanes 0-15) or 8+v (lanes 16-31); N = lane&15
    const float bv0 = b[ 0 + nlane];
    const float bv1 = b[16 + nlane];
    const float bv2 = b[32 + nlane];
    const float bv3 = b[48 + nlane];
    const int   rbase = row0 + ((lane >> 4) << 3);

    #pragma unroll
    for (int v = 0; v < 8; ++v) {
        float* hr = h + (size_t)(rbase + v) * OUT_FEATS;
        hr[ 0 + nlane] = fmaxf(c0[v] + bv0, 0.0f);
        hr[16 + nlane] = fmaxf(c1[v] + bv1, 0.0f);
        hr[32 + nlane] = fmaxf(c2[v] + bv2, 0.0f);
        hr[48 + nlane] = fmaxf(c3[v] + bv3, 0.0f);
    }
}

// ---------------- edge gather + scatter-add (16 threads per edge) ----------------
__global__ void GNN_scatter_kernel(const float* __restrict__ h,
                                   const int* __restrict__ edge_src,
                                   const int* __restrict__ edge_dst,
                                   float* __restrict__ out) {
    long long t = (long long)blockIdx.x * blockDim.x + threadIdx.x;
    int e = (int)(t >> 4);                 // edge id
    if (e >= N_EDGES) return;
    int q = (int)(t & 15);                 // 4-float chunk within the 64 features
    int s = edge_src[e];
    int d = edge_dst[e];
    const float4 v = *(const float4*)(h + (size_t)s * OUT_FEATS + q * 4);
    float* o = out + (size_t)d * OUT_FEATS + q * 4;
    atomicAdd(o + 0, v.x);
    atomicAdd(o + 1, v.y);
    atomicAdd(o + 2, v.z);
    atomicAdd(o + 3, v.w);
}

// ---------------- out /= max(deg, 1) ----------------
__global__ void GNN_finalize_kernel(float* __restrict__ out, const float* __restrict__ deg) {
    int i = blockIdx.x * blockDim.x + threadIdx.x;
    if (i >= N_NODES * OUT_FEATS) return;
    float dg = fmaxf(deg[i >> 6], 1.0f);
    out[i] = out[i] / dg;
}

extern "C" void kernel_launch(void* const* d_in, const int* in_sizes, int n_in,
                              void* d_out, int out_size, void* d_ws, size_t ws_size,
                              hipStream_t stream) {
    const float* x    = (const float*)d_in[0];   // [50000, 256]
    const float* W    = (const float*)d_in[1];   // [64, 256]
    const float* b    = (const float*)d_in[2];   // [64]
    const int*   esrc = (const int*)d_in[3];     // [1.6M]
    const int*   edst = (const int*)d_in[4];     // [1.6M]
    float*       out  = (float*)d_out;           // [50000, 64]

    float* h   = (float*)d_ws;                       // 12.8 MB
    float* deg = h + (size_t)N_NODES * OUT_FEATS;    // 200 KB

    const int ZERO_N = N_NODES * OUT_FEATS;
    GNN_zero_kernel<<<(ZERO_N + 255) / 256, 256, 0, stream>>>(out, deg);
    GNN_deg_kernel<<<(N_EDGES + 255) / 256, 256, 0, stream>>>(edst, deg);
    GNN_gemm_relu_kernel<<<(M_TILES + 7) / 8, 256, 0, stream>>>(x, W, b, h);
    long long sc_threads = (long long)N_EDGES * 16;
    GNN_scatter_kernel<<<(unsigned)((sc_threads + 255) / 256), 256, 0, stream>>>(h, esrc, edst, out);
    GNN_finalize_kernel<<<(ZERO_N + 255) / 256, 256, 0, stream>>>(out, deg);
}